// deform_conv_model_44616120270871
// MI455X (gfx1250) — compile-verified
//
#include <hip/hip_runtime.h>

typedef __attribute__((ext_vector_type(2))) float v2f;
typedef __attribute__((ext_vector_type(8))) float v8f;

#define B_   8
#define C_   6
#define O_   10
#define KW_  8
#define K_   56
#define PK_  57       /* padded row stride for per-(pos,k) LDS arrays */
#define OH_  128
#define OW_  128
#define H_   134
#define W_   135
#define CK_  336      /* K_*C_, flatten ck = k*C_ + c */
#define CHUNKS_ 84    /* CK_/4 */

__global__ __launch_bounds__(64) void dcn_wmma_kernel(
    const float* __restrict__ x, const float* __restrict__ offset,
    const float* __restrict__ mask, const float* __restrict__ weight,
    float* __restrict__ out)
{
  // per-(pos,k): 4 folded bilinear weights + 4 packed 16-bit element indices
  __shared__ float4 s_w4[32 * PK_];          // 29184 B
  __shared__ uint2  s_i2[32 * PK_];          // 14592 B
  // weight pre-packed for B fragments: s_b[ck/2][n*2 + (ck&1)]
  __shared__ float  s_b[CK_ / 2][32];        // 21504 B   (total 65280 B)

  const int tid = threadIdx.x;
  const int ow0 = blockIdx.x * 32;
  const int oh  = blockIdx.y;
  const int b   = blockIdx.z;

  // ---- phase 0a: coalesced weight load (native linear order), scatter to LDS
  for (int idx = tid; idx < CK_ * O_; idx += 64) {   // 3360 elements
    int o = idx / CK_;
    int r = idx - o * CK_;          // c*K_ + k
    int c = r / K_;
    int k = r - c * K_;
    int ck = k * C_ + c;
    s_b[ck >> 1][o * 2 + (ck & 1)] = weight[idx];
  }
  // ---- phase 0b: zero-fill N padding (out channels 10..15)
  for (int idx = tid; idx < CK_ * 6; idx += 64) {
    int ck = idx / 6;
    int n  = O_ + (idx - ck * 6);
    s_b[ck >> 1][n * 2 + (ck & 1)] = 0.0f;
  }

  // ---- phase 1: bilinear setup once per (position, tap); pos fastest -> coalesced
  for (int e = tid; e < 32 * K_; e += 64) {
    int pos = e & 31;
    int k   = e >> 5;
    int ow  = ow0 + pos;
    int ki = k >> 3, kj = k & 7;               // KW_ == 8
    float offy = offset[(((b * 2 * K_) + 2 * k) * OH_ + oh) * OW_ + ow];
    float offx = offset[(((b * 2 * K_) + 2 * k + 1) * OH_ + oh) * OW_ + ow];
    float m    = mask[((b * K_ + k) * OH_ + oh) * OW_ + ow];
    float py = (float)(oh + ki) + offy;
    float px = (float)(ow + kj) + offx;
    float fy = floorf(py), fx = floorf(px);
    float wy1 = py - fy, wx1 = px - fx;
    float wy0 = 1.0f - wy1, wx0 = 1.0f - wx1;
    int y0 = (int)fy, x0 = (int)fx;
    int y1 = y0 + 1,  x1 = x0 + 1;
    float vy0 = (y0 >= 0 && y0 < H_) ? 1.0f : 0.0f;
    float vy1 = (y1 >= 0 && y1 < H_) ? 1.0f : 0.0f;
    float vx0 = (x0 >= 0 && x0 < W_) ? 1.0f : 0.0f;
    float vx1 = (x1 >= 0 && x1 < W_) ? 1.0f : 0.0f;
    int y0c = min(max(y0, 0), H_ - 1), y1c = min(max(y1, 0), H_ - 1);
    int x0c = min(max(x0, 0), W_ - 1), x1c = min(max(x1, 0), W_ - 1);
    float4 wt;
    wt.x = wy0 * wx0 * m * vy0 * vx0;
    wt.y = wy0 * wx1 * m * vy0 * vx1;
    wt.z = wy1 * wx0 * m * vy1 * vx0;
    wt.w = wy1 * wx1 * m * vy1 * vx1;
    unsigned i00 = (unsigned)(y0c * W_ + x0c);
    unsigned i01 = (unsigned)(y0c * W_ + x1c);
    unsigned i10 = (unsigned)(y1c * W_ + x0c);
    unsigned i11 = (unsigned)(y1c * W_ + x1c);
    int slot = pos * PK_ + k;
    s_w4[slot] = wt;
    s_i2[slot] = make_uint2(i00 | (i01 << 16), i10 | (i11 << 16));
  }
  __syncthreads();

  // ---- phase 2: 84 x V_WMMA_F32_16X16X4_F32 per wave (16 positions x 16 outs)
  const int lane = tid & 31;
  const int wv   = tid >> 5;       // wave in block: positions wv*16..wv*16+15
  const int h    = lane >> 4;      // lane half selects K slots {2h, 2h+1}
  const int n16  = lane & 15;      // A row (position) and B column (out chan)
  const int p    = wv * 16 + n16;  // local position this lane feeds into A
  const float* xb = x + (size_t)b * C_ * H_ * W_;

  v8f acc = {0.f, 0.f, 0.f, 0.f, 0.f, 0.f, 0.f, 0.f};

  for (int cc = 0; cc < CHUNKS_; ++cc) {
    int ck0 = cc * 4 + h * 2;            // even -> c0 even -> pair shares tap k
    int k  = ck0 / C_;
    int c0 = ck0 - k * C_;
    int slot = p * PK_ + k;
    float4 wt = s_w4[slot];
    uint2  pi = s_i2[slot];
    unsigned i00 = pi.x & 0xffffu, i01 = pi.x >> 16;
    unsigned i10 = pi.y & 0xffffu, i11 = pi.y >> 16;
    const float* xc0 = xb + c0 * (H_ * W_);
    const float* xc1 = xc0 + (H_ * W_);
    float a0 = wt.x * xc0[i00] + wt.y * xc0[i01] + wt.z * xc0[i10] + wt.w * xc0[i11];
    float a1 = wt.x * xc1[i00] + wt.y * xc1[i01] + wt.z * xc1[i10] + wt.w * xc1[i11];
    v2f A  = {a0, a1};
    v2f Bf = *(const v2f*)&s_b[ck0 >> 1][n16 * 2];
    acc = __builtin_amdgcn_wmma_f32_16x16x4_f32(
        /*neg_a=*/false, A, /*neg_b=*/false, Bf,
        /*c_mod=*/(short)0, acc, /*reuse_a=*/false, /*reuse_b=*/false);
  }

  // D layout: lane half h, VGPR r holds D[M = r + 8h][N = n16]
  if (n16 < O_) {
    int owBase = ow0 + wv * 16 + h * 8;
    float* op = out + (((size_t)b * O_ + n16) * OH_ + oh) * OW_ + owBase;
    float4 lo = {acc[0], acc[1], acc[2], acc[3]};
    float4 hi = {acc[4], acc[5], acc[6], acc[7]};
    *(float4*)(op)     = lo;
    *(float4*)(op + 4) = hi;
  }
}

extern "C" void kernel_launch(void* const* d_in, const int* in_sizes, int n_in,
                              void* d_out, int out_size, void* d_ws, size_t ws_size,
                              hipStream_t stream) {
  (void)in_sizes; (void)n_in; (void)out_size; (void)d_ws; (void)ws_size;
  const float* x      = (const float*)d_in[0];
  const float* offset = (const float*)d_in[1];
  const float* mask   = (const float*)d_in[2];
  const float* weight = (const float*)d_in[3];
  float* out = (float*)d_out;
  dim3 grid(OW_ / 32, OH_, B_);
  dcn_wmma_kernel<<<grid, dim3(64), 0, stream>>>(x, offset, mask, weight, out);
}